// CombinedLoss_14860586844749
// MI455X (gfx1250) — compile-verified
//
#include <hip/hip_runtime.h>

// CombinedLoss for MI455X (gfx1250, wave32).
//
// Math collapse: dist_maps[:,1] == (argmax(probs,axis=1) != 1) exactly
// (see analysis), so the whole loss is one streaming pass accumulating:
//   surf = sum over voxels of p1 * [argmax != 1]
//   tp   = sum p*t ; fp = sum (1-t)*p ; fn = sum t*(1-p)
// loss = surf/(B*D*H*W) + 1 - (tp+1)/(tp + 0.5*fp + 0.5*fn + 1)
// Memory bound: 50.3 MB read / 23.3 TB/s ~ 2.2 us.
//
// Block reductions use V_WMMA_F32_16X16X4_F32 (f32 in/out, exact):
// A = data (documented 16x4 layout), B = all-ones (layout-immune),
// C chained accumulator -> D[i][j] = sum_k A[i][k] + C[i][j].

typedef __attribute__((ext_vector_type(2))) float v2f;
typedef __attribute__((ext_vector_type(8))) float v8f;

#define TPB 256
#define MAXNB 1024

// Deterministic 4-component block reduction via WMMA.
// Each of TPB threads contributes (s0..s3); result in out[0..3], valid on
// all lanes of wave 0. lds must hold 4*TPB floats.
__device__ __forceinline__ void block_reduce4_wmma(float s0, float s1, float s2,
                                                   float s3, float* lds,
                                                   float out[4]) {
    const int tid = threadIdx.x;
    // component-major staging: comp c occupies lds[c*TPB .. c*TPB+TPB-1]
    lds[0 * TPB + tid] = s0;
    lds[1 * TPB + tid] = s1;
    lds[2 * TPB + tid] = s2;
    lds[3 * TPB + tid] = s3;
    __syncthreads();
    if (tid < 32) {  // wave 0 only; EXEC all-ones inside this branch
        const int lane = tid;
        const int comp = lane & 3;  // row M = lane%16, M%4 == lane%4
        const int rank = ((lane & 15) >> 2) + ((lane >> 4) << 2);  // 0..7
        const float* src = lds + comp * TPB + rank * 2;
        v8f acc = {0.f, 0.f, 0.f, 0.f, 0.f, 0.f, 0.f, 0.f};
        v2f bones;
        bones.x = 1.0f;
        bones.y = 1.0f;  // B = all ones (every element), layout-immune
#pragma unroll
        for (int it = 0; it < TPB / 16; ++it) {
            // A 16x4 f32 layout: lanes 0-15 -> K=0,1 ; lanes 16-31 -> K=2,3
            v2f a;
            a.x = src[it * 16 + 0];
            a.y = src[it * 16 + 1];
            acc = __builtin_amdgcn_wmma_f32_16x16x4_f32(
                false, a, false, bones, (short)0, acc, false, false);
        }
        // D[i][j] = sum_k A[i][k] (all columns equal). Component c lives in
        // rows {c, c+4, c+8, c+12}. D layout: lane t<16 holds D[v][t] in
        // acc[v]; lane t>=16 holds D[v+8][t-16].
        float p0 = acc[0] + acc[4];
        float p1 = acc[1] + acc[5];
        float p2 = acc[2] + acc[6];
        float p3 = acc[3] + acc[7];
        out[0] = __shfl(p0, 0) + __shfl(p0, 16);
        out[1] = __shfl(p1, 0) + __shfl(p1, 16);
        out[2] = __shfl(p2, 0) + __shfl(p2, 16);
        out[3] = __shfl(p3, 0) + __shfl(p3, 16);
    }
}

__device__ __forceinline__ void voxel_accum(float pa, float pb, float pc,
                                            float ta, float tb, float tc,
                                            float& surf, float& tp, float& fp,
                                            float& fn) {
    // argmax == 1 iff (p1 > p0) && (p1 >= p2)  (first-max tie rule)
    surf += (pb > pa && pb >= pc) ? 0.0f : pb;
    tp += pa * ta + pb * tb + pc * tc;
    fp += (1.0f - ta) * pa + (1.0f - tb) * pb + (1.0f - tc) * pc;
    fn += ta * (1.0f - pa) + tb * (1.0f - pb) + tc * (1.0f - pc);
}

__global__ void loss_partial_kernel(const float* __restrict__ probs,
                                    const float* __restrict__ target,
                                    float4* __restrict__ blockOut, int nGroups,
                                    int S /* channel stride in floats */) {
    __shared__ float lds[4 * TPB];
    float surf = 0.f, tp = 0.f, fp = 0.f, fn = 0.f;
    const int stride = gridDim.x * blockDim.x;
    for (int g = blockIdx.x * blockDim.x + threadIdx.x; g < nGroups;
         g += stride) {
        const int n = g * 4;     // voxel index (4 voxels per group)
        const int b = n / S;     // S is a power of two -> shift
        const int base = n + 2 * S * b;  // (b*3 + 0)*S + s
        const float4 p0 = *(const float4*)(probs + base);
        const float4 p1 = *(const float4*)(probs + base + S);
        const float4 p2 = *(const float4*)(probs + base + 2 * S);
        const float4 t0 = *(const float4*)(target + base);
        const float4 t1 = *(const float4*)(target + base + S);
        const float4 t2 = *(const float4*)(target + base + 2 * S);
        voxel_accum(p0.x, p1.x, p2.x, t0.x, t1.x, t2.x, surf, tp, fp, fn);
        voxel_accum(p0.y, p1.y, p2.y, t0.y, t1.y, t2.y, surf, tp, fp, fn);
        voxel_accum(p0.z, p1.z, p2.z, t0.z, t1.z, t2.z, surf, tp, fp, fn);
        voxel_accum(p0.w, p1.w, p2.w, t0.w, t1.w, t2.w, surf, tp, fp, fn);
    }
    float o[4];
    block_reduce4_wmma(surf, tp, fp, fn, lds, o);
    if (threadIdx.x == 0) {
        float4 r;
        r.x = o[0];
        r.y = o[1];
        r.z = o[2];
        r.w = o[3];
        blockOut[blockIdx.x] = r;
    }
}

__global__ void loss_final_kernel(const float4* __restrict__ blockIn, int nb,
                                  float* __restrict__ out, float invMean) {
    __shared__ float lds[4 * TPB];
    float s0 = 0.f, s1 = 0.f, s2 = 0.f, s3 = 0.f;
    for (int i = threadIdx.x; i < nb; i += TPB) {
        const float4 v = blockIn[i];
        s0 += v.x;
        s1 += v.y;
        s2 += v.z;
        s3 += v.w;
    }
    float o[4];
    block_reduce4_wmma(s0, s1, s2, s3, lds, o);
    if (threadIdx.x == 0) {
        const float surface = o[0] * invMean;  // invMean = 2^-21, exact
        const float tp = o[1], fp = o[2], fn = o[3];
        const float tversky =
            1.0f - (tp + 1.0f) / (tp + 0.5f * fp + 0.5f * fn + 1.0f);
        out[0] = surface + tversky;
    }
}

extern "C" void kernel_launch(void* const* d_in, const int* in_sizes, int n_in,
                              void* d_out, int out_size, void* d_ws,
                              size_t ws_size, hipStream_t stream) {
    const float* probs = (const float*)d_in[0];
    const float* target = (const float*)d_in[1];
    const int S = 64 * 128 * 128;  // channel stride (D*H*W)
    const int V = 2 * S;           // B*D*H*W voxels
    const int nGroups = V / 4;

    int nb = MAXNB;
    while (nb > 1 && (size_t)nb * sizeof(float4) > ws_size) nb >>= 1;
    float4* bpart = (float4*)d_ws;

    loss_partial_kernel<<<nb, TPB, 0, stream>>>(probs, target, bpart, nGroups,
                                                S);
    loss_final_kernel<<<1, TPB, 0, stream>>>(bpart, nb, (float*)d_out,
                                             1.0f / (float)V);
}